// DenseFlashAttention_58712202936473
// MI455X (gfx1250) — compile-verified
//
#include <hip/hip_runtime.h>
#include <hip/hip_bf16.h>
#include <math.h>

typedef __attribute__((ext_vector_type(2))) float v2f;
typedef __attribute__((ext_vector_type(8))) float v8f;

#define N_NODES 20000
#define N_EDGES 320000
#define FEAT    64
#define HEADS   4

// ---------------- workspace layout (float units) ----------------
#define SZ_PROJ      (3ull * HEADS * N_NODES * FEAT)   // 12 projection matrices
#define SZ_LOGITS    ((unsigned long long)HEADS * N_EDGES)
#define SZ_SEG       ((unsigned long long)N_NODES * HEADS)
#define SZ_ACCUM     ((unsigned long long)N_NODES * FEAT)

#define OFF_PROJ     0ull
#define OFF_REXP     (OFF_PROJ + SZ_PROJ)
#define OFF_TEXP     (OFF_REXP + SZ_LOGITS)
#define OFF_SEGMAX_R (OFF_TEXP + SZ_LOGITS)
#define OFF_SEGMAX_T (OFF_SEGMAX_R + SZ_SEG)
#define OFF_DENOM_R  (OFF_SEGMAX_T + SZ_SEG)
#define OFF_DENOM_T  (OFF_DENOM_R + SZ_SEG)
#define OFF_ACCUM    (OFF_DENOM_T + SZ_SEG)
#define WS_FLOATS    (OFF_ACCUM + SZ_ACCUM)

// ---------------- helpers ----------------
__device__ __forceinline__ float softplusf(float v) {
  return v > 20.f ? v : log1pf(__expf(v));
}
// monotone float<->uint mapping for atomicMax on floats
__device__ __forceinline__ unsigned int fkey(float f) {
  unsigned int u = __float_as_uint(f);
  return (u & 0x80000000u) ? ~u : (u | 0x80000000u);
}
__device__ __forceinline__ float funkey(unsigned int k) {
  unsigned int u = (k & 0x80000000u) ? (k ^ 0x80000000u) : ~k;
  return __uint_as_float(u);
}

// ---------------- kernel 0: zero segment/accumulator region ----------------
__global__ void gatk_init(float* __restrict__ ws) {
  size_t i = (size_t)blockIdx.x * blockDim.x + threadIdx.x;
  size_t n = WS_FLOATS - OFF_SEGMAX_R;
  if (i < n) ws[OFF_SEGMAX_R + i] = 0.0f;   // 0 bits == smallest fkey and float 0
}

// ---------------- kernel 1: 12 projection GEMMs via f32 WMMA ----------------
// proj[m][n][g] = sum_f x[n][f] * W_m[f][g],  m = {proj,radial,tang} x 4 heads
__global__ void gatk_proj_wmma(const float* __restrict__ x,
                               const float* __restrict__ wp,
                               const float* __restrict__ wr,
                               const float* __restrict__ wt,
                               float* __restrict__ proj) {
  const int lane = threadIdx.x & 31;
  const int task = blockIdx.x * 8 + (threadIdx.x >> 5);
  const int rowTiles = N_NODES / 16;       // 1250
  const int colTiles = FEAT / 16;          // 4
  const int tasksPerMat = rowTiles * colTiles;
  if (task >= 12 * tasksPerMat) return;    // exact multiple; never divergent
  const int mat  = task / tasksPerMat;
  const int rem  = task - mat * tasksPerMat;
  const int row0 = (rem / colTiles) * 16;
  const int n0   = (rem % colTiles) * 16;
  const int m3   = mat >> 2, h = mat & 3;
  const float* W = (m3 == 0 ? wp : (m3 == 1 ? wr : wt)) + (size_t)h * FEAT * FEAT;

  v8f c = {};
  const int aRow = row0 + (lane & 15);
  const int kOff = 2 * (lane >> 4);        // lanes 16-31 hold K+2,K+3
  const int ln   = lane & 15;
#pragma unroll
  for (int kk = 0; kk < FEAT / 4; ++kk) {
    const int k0 = kk * 4;
    v2f a = *(const v2f*)(x + (size_t)aRow * FEAT + k0 + kOff);
    v2f b;
    const int kb = k0 + kOff;
    b.x = W[(size_t)kb * FEAT + n0 + ln];
    b.y = W[(size_t)(kb + 1) * FEAT + n0 + ln];
    c = __builtin_amdgcn_wmma_f32_16x16x4_f32(false, a, false, b, (short)0, c,
                                              false, false);
  }
  float* dst = proj + ((size_t)mat * N_NODES + row0) * FEAT + n0 + ln;
  const int mhi = (lane >> 4) * 8;
#pragma unroll
  for (int v = 0; v < 8; ++v)
    dst[(size_t)(v + mhi) * FEAT] = c[v];
}

// ---------------- kernel 2: per-edge logits + segment max (wave per edge) ---
__global__ void gatk_edge_logits(const float* __restrict__ proj,
                                 const int* __restrict__ ei,
                                 const float* __restrict__ elen,
                                 const float* __restrict__ rscore,
                                 const float* __restrict__ tscore,
                                 const float* __restrict__ dls,
                                 const float* __restrict__ rtb,
                                 const float* __restrict__ rtw,
                                 float* __restrict__ rlog,
                                 float* __restrict__ tlog,
                                 unsigned int* __restrict__ segr,
                                 unsigned int* __restrict__ segt) {
  const int lane = threadIdx.x & 31;
  const int e = blockIdx.x * 8 + (threadIdx.x >> 5);
  if (e >= N_EDGES) return;
  const int s = ei[e];
  const int r = ei[N_EDGES + e];
  const float len = elen[e];
  const float dist_scale = softplusf(dls[0]);
  const int f0 = lane * 2;
#pragma unroll
  for (int h = 0; h < HEADS; ++h) {
    const float* ps = proj + ((size_t)h * N_NODES + s) * FEAT + f0;
    const float* pr = proj + ((size_t)h * N_NODES + r) * FEAT + f0;
    const float d0 = ps[0] - pr[0], d1 = ps[1] - pr[1];
    float sr = d0 * rscore[h * FEAT + f0] + d1 * rscore[h * FEAT + f0 + 1];
    float st = d0 * tscore[h * FEAT + f0] + d1 * tscore[h * FEAT + f0 + 1];
    for (int off = 16; off; off >>= 1) {
      sr += __shfl_xor(sr, off, 32);
      st += __shfl_xor(st, off, 32);
    }
    if (lane == 0) {
      float rl = sr - dist_scale * len;
      const float temp = softplusf(rtb[h] + rtw[h] * len);
      rl = rl / (temp + 1e-4f);
      rlog[(size_t)h * N_EDGES + e] = rl;
      tlog[(size_t)h * N_EDGES + e] = st;
      atomicMax(segr + (size_t)r * HEADS + h, fkey(rl));
      atomicMax(segt + (size_t)r * HEADS + h, fkey(st));
    }
  }
}

// ---------------- kernel 3: exp + segment denominator (thread per edge) ----
__global__ void gatk_edge_exp(const int* __restrict__ ei,
                              float* __restrict__ rlog,
                              float* __restrict__ tlog,
                              const unsigned int* __restrict__ segr,
                              const unsigned int* __restrict__ segt,
                              float* __restrict__ denr,
                              float* __restrict__ dent) {
  const int e = blockIdx.x * blockDim.x + threadIdx.x;
  if (e >= N_EDGES) return;
  const int r = ei[N_EDGES + e];
#pragma unroll
  for (int h = 0; h < HEADS; ++h) {
    const size_t li = (size_t)h * N_EDGES + e;
    const size_t si = (size_t)r * HEADS + h;
    const float exr = __expf(rlog[li] - funkey(segr[si]));
    rlog[li] = exr;                       // reuse logit buffer for exp values
    atomicAdd(denr + si, exr);
    const float ext = __expf(tlog[li] - funkey(segt[si]));
    tlog[li] = ext;
    atomicAdd(dent + si, ext);
  }
}

// ---------------- kernel 4: blended weighted scatter (wave per edge) -------
__global__ void gatk_edge_scatter(const float* __restrict__ proj,
                                  const int* __restrict__ ei,
                                  const float* __restrict__ elen,
                                  const float* __restrict__ rexp,
                                  const float* __restrict__ texp,
                                  const float* __restrict__ denr,
                                  const float* __restrict__ dent,
                                  const float* __restrict__ mixb,
                                  const float* __restrict__ mixs,
                                  float* __restrict__ accum) {
  const int lane = threadIdx.x & 31;
  const int e = blockIdx.x * 8 + (threadIdx.x >> 5);
  if (e >= N_EDGES) return;
  const int s = ei[e], r = ei[N_EDGES + e];
  const float len = elen[e];
  const float* rproj = proj + (size_t)HEADS * N_NODES * FEAT;
  const float* tproj = proj + 2ull * HEADS * N_NODES * FEAT;
  const int f0 = lane * 2;
  float a0 = 0.f, a1 = 0.f;
#pragma unroll
  for (int h = 0; h < HEADS; ++h) {
    const float gate = 1.f / (1.f + __expf(-(mixb[h] + mixs[h] * len)));
    const size_t li = (size_t)h * N_EDGES + e;
    const size_t si = (size_t)r * HEADS + h;
    const float ra = rexp[li] / (denr[si] + 1e-9f);
    const float ta = texp[li] / (dent[si] + 1e-9f);
    const float ba = gate * ra + (1.f - gate) * ta;
    const float* rs = rproj + ((size_t)h * N_NODES + s) * FEAT + f0;
    const float* rr = rproj + ((size_t)h * N_NODES + r) * FEAT + f0;
    const float* ts = tproj + ((size_t)h * N_NODES + s) * FEAT + f0;
    const float* tr = tproj + ((size_t)h * N_NODES + r) * FEAT + f0;
    const float rd0 = rs[0] - rr[0], rd1 = rs[1] - rr[1];
    const float td0 = ts[0] - tr[0], td1 = ts[1] - tr[1];
    a0 += ba * (gate * rd0 + (1.f - gate) * td0);
    a1 += ba * (gate * rd1 + (1.f - gate) * td1);
  }
  a0 *= (1.f / HEADS);
  a1 *= (1.f / HEADS);
  atomicAdd(accum + (size_t)r * FEAT + f0, a0);
  atomicAdd(accum + (size_t)r * FEAT + f0 + 1, a1);
}

// ---------------- kernel 5: out = x + accum @ w_out (f32 WMMA) -------------
__global__ void gatk_out_wmma(const float* __restrict__ accum,
                              const float* __restrict__ x,
                              const float* __restrict__ wout,
                              float* __restrict__ out) {
  const int lane = threadIdx.x & 31;
  const int task = blockIdx.x * 8 + (threadIdx.x >> 5);
  const int rowTiles = N_NODES / 16, colTiles = FEAT / 16;
  if (task >= rowTiles * colTiles) return; // exact multiple; never divergent
  const int row0 = (task / colTiles) * 16;
  const int n0   = (task % colTiles) * 16;
  v8f c = {};
  const int aRow = row0 + (lane & 15);
  const int kOff = 2 * (lane >> 4);
  const int ln   = lane & 15;
#pragma unroll
  for (int kk = 0; kk < FEAT / 4; ++kk) {
    const int k0 = kk * 4;
    v2f a = *(const v2f*)(accum + (size_t)aRow * FEAT + k0 + kOff);
    v2f b;
    const int kb = k0 + kOff;
    b.x = wout[(size_t)kb * FEAT + n0 + ln];
    b.y = wout[(size_t)(kb + 1) * FEAT + n0 + ln];
    c = __builtin_amdgcn_wmma_f32_16x16x4_f32(false, a, false, b, (short)0, c,
                                              false, false);
  }
  const int mhi = (lane >> 4) * 8;
#pragma unroll
  for (int v = 0; v < 8; ++v) {
    const size_t idx = (size_t)(row0 + v + mhi) * FEAT + n0 + ln;
    out[idx] = x[idx] + c[v];
  }
}

// ---------------- launch ----------------
extern "C" void kernel_launch(void* const* d_in, const int* in_sizes, int n_in,
                              void* d_out, int out_size, void* d_ws, size_t ws_size,
                              hipStream_t stream) {
  (void)in_sizes; (void)n_in; (void)out_size; (void)ws_size;
  const float* x      = (const float*)d_in[0];
  const int*   ei     = (const int*)d_in[1];     // [2, E]
  // d_in[2] = edge_vec (unused by reference)
  const float* elen   = (const float*)d_in[3];
  const float* wp     = (const float*)d_in[4];
  const float* wr     = (const float*)d_in[5];
  const float* wt     = (const float*)d_in[6];
  const float* rsc    = (const float*)d_in[7];
  const float* tsc    = (const float*)d_in[8];
  const float* dls    = (const float*)d_in[9];
  const float* rtb    = (const float*)d_in[10];
  const float* rtw    = (const float*)d_in[11];
  const float* mixb   = (const float*)d_in[12];
  const float* mixs   = (const float*)d_in[13];
  const float* wout   = (const float*)d_in[14];
  float* out = (float*)d_out;

  float* ws = (float*)d_ws;
  float* proj        = ws + OFF_PROJ;
  float* rlog        = ws + OFF_REXP;
  float* tlog        = ws + OFF_TEXP;
  unsigned int* segr = (unsigned int*)(ws + OFF_SEGMAX_R);
  unsigned int* segt = (unsigned int*)(ws + OFF_SEGMAX_T);
  float* denr        = ws + OFF_DENOM_R;
  float* dent        = ws + OFF_DENOM_T;
  float* accum       = ws + OFF_ACCUM;

  // 0) zero segment max/denominator/accumulator region
  {
    const size_t n = WS_FLOATS - OFF_SEGMAX_R;
    gatk_init<<<(unsigned)((n + 255) / 256), 256, 0, stream>>>(ws);
  }
  // 1) 12 projection GEMMs: 12 * 1250 * 4 = 60000 wave-tiles, 8 waves/block
  gatk_proj_wmma<<<60000 / 8, 256, 0, stream>>>(x, wp, wr, wt, proj);
  // 2) edge logits + segment max: one wave per edge
  gatk_edge_logits<<<N_EDGES / 8, 256, 0, stream>>>(proj, ei, elen, rsc, tsc,
                                                    dls, rtb, rtw, rlog, tlog,
                                                    segr, segt);
  // 3) exp + denominators: one thread per edge
  gatk_edge_exp<<<N_EDGES / 256, 256, 0, stream>>>(ei, rlog, tlog, segr, segt,
                                                   denr, dent);
  // 4) blended weighted scatter: one wave per edge
  gatk_edge_scatter<<<N_EDGES / 8, 256, 0, stream>>>(proj, ei, elen, rlog, tlog,
                                                     denr, dent, mixb, mixs,
                                                     accum);
  // 5) output GEMM + residual: 1250 * 4 = 5000 wave-tiles
  gatk_out_wmma<<<5000 / 8, 256, 0, stream>>>(accum, x, wout, out);
}